// DCM_29394756174378
// MI455X (gfx1250) — compile-verified
//
#include <hip/hip_runtime.h>

// ---------------- problem constants ----------------
#define B_      16
#define C_      128
#define H_      128
#define W_      128
#define PLANES  (B_ * C_)      // 2048
#define PSIZE   (H_ * W_)      // 16384

// LDS plane layout: rows -1..128 (130), pitch 136 floats, interior starts at col 4
// -> every interior row start is 16B aligned (544*(r+1)+16), halo at cols 3 and 132.
#define PITCH   136
#define CO      4
#define NROWS   130

__device__ __forceinline__ int lidx(int r, int c) { return (r + 1) * PITCH + (c + CO); }

// ---------------- CDNA5 async global->LDS path ----------------
typedef __attribute__((ext_vector_type(4))) int v4i;
typedef v4i __attribute__((address_space(1)))* gv4i_p;  // global
typedef v4i __attribute__((address_space(3)))* lv4i_p;  // LDS

#if defined(__AMDGCN__) && __has_builtin(__builtin_amdgcn_global_load_async_to_lds_b128)
#define USE_ASYNC_LDS 1
#else
#define USE_ASYNC_LDS 0
#endif

__device__ __forceinline__ void wait_async_lds() {
#if defined(__AMDGCN__)
#if __has_builtin(__builtin_amdgcn_s_wait_asynccnt)
  __builtin_amdgcn_s_wait_asynccnt(0);
#else
  asm volatile("s_wait_asynccnt 0" ::: "memory");
#endif
#endif
}

// Load one 128x128 f32 plane into padded LDS (zero halo). 256 threads.
__device__ __forceinline__ void load_plane(const float* __restrict__ xp,
                                           float* __restrict__ splane, int tid) {
  // zero halo: top/bottom rows (c = -1..128) and left/right cols (r = 0..127)
  for (int i = tid; i < NROWS; i += 256) {
    splane[lidx(-1, i - 1)]  = 0.0f;
    splane[lidx(128, i - 1)] = 0.0f;
  }
  for (int r = tid; r < 128; r += 256) {
    splane[lidx(r, -1)]  = 0.0f;
    splane[lidx(r, 128)] = 0.0f;
  }
  // interior: 128 rows x 512B; each thread moves one 16B chunk per row-group
  const int lane16 = tid & 31;   // 16B chunk within row
  const int rg     = tid >> 5;   // 8 row groups
  for (int r = rg; r < 128; r += 8) {
    const float* g = xp + r * W_ + lane16 * 4;
    float*       l = &splane[lidx(r, lane16 * 4)];
#if USE_ASYNC_LDS
    __builtin_amdgcn_global_load_async_to_lds_b128(
        (gv4i_p)(void*)g, (lv4i_p)l, 0, 0);
#else
    const float4 v = *(const float4*)g;
    *(float4*)l = v;
#endif
  }
#if USE_ASYNC_LDS
  wait_async_lds();
#endif
  __syncthreads();
}

// Adaptive 3x3 avg-pool for H=W=128: bins [0,43),[43,86),[86,128). Result in sf[9].
__device__ __forceinline__ void compute_filter(const float* __restrict__ splane,
                                               float* __restrict__ scol,
                                               float* __restrict__ sf, int tid) {
  if (tid < 128) {
    float a0 = 0.f, a1 = 0.f, a2 = 0.f;
    for (int r = 0;  r < 43;  ++r) a0 += splane[lidx(r, tid)];
    for (int r = 43; r < 86;  ++r) a1 += splane[lidx(r, tid)];
    for (int r = 86; r < 128; ++r) a2 += splane[lidx(r, tid)];
    scol[0 * 128 + tid] = a0;
    scol[1 * 128 + tid] = a1;
    scol[2 * 128 + tid] = a2;
  }
  __syncthreads();
  if (tid < 9) {
    const int   sw[3]  = {0, 43, 86};
    const int   ew[3]  = {43, 86, 128};
    const float cnt[3] = {43.f, 43.f, 42.f};
    const int bi = tid / 3, bj = tid % 3;
    float s = 0.f;
    for (int c = sw[bj]; c < ew[bj]; ++c) s += scol[bi * 128 + c];
    sf[tid] = s / (cnt[bi] * cnt[bj]);
  }
  __syncthreads();
}

// One block per (b,c) plane. WRITE_OUT=false: emit per-plane (sum,sumsq).
// WRITE_OUT=true: apply per-channel scale/bias + ReLU, write output (non-temporal).
template <bool WRITE_OUT>
__global__ __launch_bounds__(256) void dcm_conv(const float* __restrict__ x,
                                                const float* __restrict__ sb,
                                                float* __restrict__ dst) {
  __shared__ float splane[NROWS * PITCH];  // ~70.7 KB
  __shared__ float scol[3 * 128];
  __shared__ float sf[9];
  __shared__ float sred[512];

  const int tid   = threadIdx.x;
  const int plane = blockIdx.x;                       // p = b*C + c
  const float* xp = x + (size_t)plane * PSIZE;

  load_plane(xp, splane, tid);
  compute_filter(splane, scol, sf, tid);

  const float f00 = sf[0], f01 = sf[1], f02 = sf[2];
  const float f10 = sf[3], f11 = sf[4], f12 = sf[5];
  const float f20 = sf[6], f21 = sf[7], f22 = sf[8];

  float scale = 0.f, bias = 0.f;
  if (WRITE_OUT) {
    const int ch = plane & (C_ - 1);
    scale = sb[ch * 2 + 0];
    bias  = sb[ch * 2 + 1];
  }

  // Each thread: one column strip of 64 rows; 3-register sliding window
  const int c  = tid & 127;
  const int r0 = (tid >> 7) * 64;

  float am = splane[lidx(r0 - 1, c - 1)], a0 = splane[lidx(r0 - 1, c)], ap = splane[lidx(r0 - 1, c + 1)];
  float bm = splane[lidx(r0,     c - 1)], b0 = splane[lidx(r0,     c)], bp = splane[lidx(r0,     c + 1)];

  float s = 0.f, s2 = 0.f;
  float* outp = WRITE_OUT ? (dst + (size_t)plane * PSIZE + (size_t)r0 * W_ + c) : nullptr;

#pragma unroll 4
  for (int r = r0; r < r0 + 64; ++r) {
    const float cm = splane[lidx(r + 1, c - 1)];
    const float c0 = splane[lidx(r + 1, c)];
    const float cp = splane[lidx(r + 1, c + 1)];
    float o = am * f00;
    o = fmaf(a0, f01, o); o = fmaf(ap, f02, o);
    o = fmaf(bm, f10, o); o = fmaf(b0, f11, o); o = fmaf(bp, f12, o);
    o = fmaf(cm, f20, o); o = fmaf(c0, f21, o); o = fmaf(cp, f22, o);
    if (WRITE_OUT) {
      float y = fmaf(o, scale, bias);
      y = fmaxf(y, 0.0f);
      __builtin_nontemporal_store(y, outp);
      outp += W_;
    } else {
      s += o;
      s2 = fmaf(o, o, s2);
    }
    am = bm; a0 = b0; ap = bp;
    bm = cm; b0 = c0; bp = cp;
  }

  if (!WRITE_OUT) {
    sred[tid]       = s;
    sred[256 + tid] = s2;
    __syncthreads();
    for (int off = 128; off > 0; off >>= 1) {
      if (tid < off) {
        sred[tid]       += sred[tid + off];
        sred[256 + tid] += sred[256 + tid + off];
      }
      __syncthreads();
    }
    if (tid == 0) {
      dst[plane * 2 + 0] = sred[0];
      dst[plane * 2 + 1] = sred[256];
    }
  }
}

// BN stats finalize: 128 threads = channels; fixed-order reduction over B=16 planes.
__global__ __launch_bounds__(128) void dcm_bn(const float* __restrict__ planeSums,
                                              const float* __restrict__ gamma,
                                              const float* __restrict__ beta,
                                              float* __restrict__ sb) {
  const int c = threadIdx.x;
  float s = 0.f, s2 = 0.f;
  for (int b = 0; b < B_; ++b) {
    const int p = b * C_ + c;
    s  += planeSums[p * 2 + 0];
    s2 += planeSums[p * 2 + 1];
  }
  const float N    = (float)B_ * (float)H_ * (float)W_;  // 262144
  const float mean = s / N;
  const float var  = s2 / N - mean * mean;
  const float inv  = rsqrtf(var + 1e-5f);
  const float scale = gamma[c] * inv;
  const float bias  = beta[c] - mean * scale;
  sb[c * 2 + 0] = scale;
  sb[c * 2 + 1] = bias;
}

extern "C" void kernel_launch(void* const* d_in, const int* in_sizes, int n_in,
                              void* d_out, int out_size, void* d_ws, size_t ws_size,
                              hipStream_t stream) {
  const float* x     = (const float*)d_in[0];
  const float* gamma = (const float*)d_in[1];
  const float* beta  = (const float*)d_in[2];
  float* out = (float*)d_out;

  float* wsf       = (float*)d_ws;
  float* planeSums = wsf;          // 2048 * 2 floats
  float* sb        = wsf + 4096;   // 128 * 2 floats

  // Pass 1: per-plane dynamic filter + conv -> per-plane sum/sumsq (fills L2 with x)
  dcm_conv<false><<<PLANES, 256, 0, stream>>>(x, nullptr, planeSums);
  // Pass 2: BN scale/bias per channel (deterministic fixed-order reduction)
  dcm_bn<<<1, 128, 0, stream>>>(planeSums, gamma, beta, sb);
  // Pass 3: recompute conv (x from L2), BN + ReLU, non-temporal store of output
  dcm_conv<true><<<PLANES, 256, 0, stream>>>(x, sb, out);
}